// IPMADGAT_7232724927270
// MI455X (gfx1250) — compile-verified
//
#include <hip/hip_runtime.h>
#include <math.h>

typedef __attribute__((ext_vector_type(2))) float v2f;
typedef __attribute__((ext_vector_type(8))) float v8f;

#define HEADS 8
#define HID 16
#define CLS 16
#define FDIM 128          // HEADS*HID == HEADS*CLS == F == 128
#define NEG_SLOPE 0.2f

// ---------------- order-preserving float<->uint map for atomic max ----------
__device__ __forceinline__ unsigned ordmap(float f) {
  unsigned u = __float_as_uint(f);
  return (u & 0x80000000u) ? ~u : (u | 0x80000000u);
}
__device__ __forceinline__ float ordunmap(unsigned u) {
  unsigned v = (u & 0x80000000u) ? (u & 0x7fffffffu) : ~u;
  return __uint_as_float(v);
}

// ordmap(-INFINITY) precomputed for host use: ~0xFF800000 = 0x007FFFFF
#define ORDMAP_NEG_INF 0x007FFFFFu

// ---------------- GEMM: H[n x 128] = X[n x 128] @ W[128 x 128] --------------
// Workgroup = 256 threads = 8 waves. W (64KB) staged once into LDS.
// Each wave computes a 16x128 output strip: 8 accumulator tiles,
// V_WMMA_F32_16X16X4_F32 over K=128, B sourced from LDS (ds_load),
// A streamed from global (one b64 per k-step).
__global__ __launch_bounds__(256) void gemm128_wmma(
    const float* __restrict__ X, const float* __restrict__ W,
    float* __restrict__ H, int n) {
  __shared__ float Wlds[FDIM * FDIM];   // 64 KB of the 320 KB WGP LDS

  // cooperative W load, float4-wide (global_load_b128 -> ds_store_b128)
  {
    const float4* __restrict__ Wv = (const float4*)W;
    float4* Wl = (float4*)Wlds;
    for (int i = threadIdx.x; i < (FDIM * FDIM) / 4; i += 256) Wl[i] = Wv[i];
  }
  __syncthreads();

  const int rowTiles = n / 16;                       // 3125
  const int rowTile = blockIdx.x * 8 + (threadIdx.x >> 5);
  if (rowTile >= rowTiles) return;                   // wave-uniform exit
  const int lane = threadIdx.x & 31;
  const int half = lane >> 4;                        // 0: lanes 0-15, 1: 16-31
  const int lm = lane & 15;
  const int m0 = rowTile * 16;

  const float* __restrict__ xrow = X + (size_t)(m0 + lm) * FDIM;

  v8f acc[8];
#pragma unroll
  for (int t = 0; t < 8; ++t) acc[t] = (v8f){};

  for (int k0 = 0; k0 < FDIM; k0 += 4) {
    const int ka = k0 + half * 2;
    v2f a;
    a.x = xrow[ka];
    a.y = xrow[ka + 1];
#pragma unroll
    for (int t = 0; t < 8; ++t) {                    // 8 column tiles
      v2f b;
      b.x = Wlds[ka * FDIM + t * 16 + lm];
      b.y = Wlds[(ka + 1) * FDIM + t * 16 + lm];
      acc[t] = __builtin_amdgcn_wmma_f32_16x16x4_f32(
          false, a, false, b, (short)0, acc[t], false, false);
    }
  }

#pragma unroll
  for (int t = 0; t < 8; ++t) {
#pragma unroll
    for (int i = 0; i < 8; ++i) {
      H[(size_t)(m0 + i + 8 * half) * FDIM + t * 16 + lm] = acc[t][i];
    }
  }
}

// ------------- attention logits: al[n,h] = sum_c H[n,h,c]*a[h,c] ------------
__global__ void attn_logits(const float* __restrict__ H,
                            const float* __restrict__ a_src,
                            const float* __restrict__ a_dst,
                            float* __restrict__ al_s,
                            float* __restrict__ al_d, int n) {
  int i = blockIdx.x * blockDim.x + threadIdx.x;  // over n*HEADS
  if (i >= n * HEADS) return;
  const int h = i & (HEADS - 1);
  const float* hp = H + (size_t)i * HID;  // [n][HEADS][HID] contiguous
  float s = 0.f, d = 0.f;
#pragma unroll
  for (int c = 0; c < HID; ++c) {
    float v = hp[c];
    s += v * a_src[h * HID + c];
    d += v * a_dst[h * HID + c];
  }
  al_s[i] = s;
  al_d[i] = d;
}

// ---------------- fills ------------------------------------------------------
__global__ void fill_u32(unsigned* __restrict__ p, unsigned v, size_t n) {
  size_t i = (size_t)blockIdx.x * blockDim.x + threadIdx.x;
  if (i < n) p[i] = v;
}
__global__ void fill_f32(float* __restrict__ p, float v, size_t n) {
  size_t i = (size_t)blockIdx.x * blockDim.x + threadIdx.x;
  if (i < n) p[i] = v;
}

// ---------------- edge helpers ----------------------------------------------
__device__ __forceinline__ void edge_sd(const int* __restrict__ src,
                                        const int* __restrict__ dst, int E,
                                        int e, int& s, int& d) {
  if (e < E) { s = src[e]; d = dst[e]; }
  else { s = e - E; d = e - E; }       // appended self loops
}

// ---------------- edge pass A: segment max of leaky(e) ----------------------
__global__ void edge_max(const int* __restrict__ src, const int* __restrict__ dst,
                         int E, int n, const float* __restrict__ al_s,
                         const float* __restrict__ al_d,
                         unsigned* __restrict__ m) {
  int e = blockIdx.x * blockDim.x + threadIdx.x;
  if (e >= E + n) return;
  int s, d;
  edge_sd(src, dst, E, e, s, d);
#pragma unroll
  for (int h = 0; h < HEADS; ++h) {
    float v = al_s[s * HEADS + h] + al_d[d * HEADS + h];
    v = v > 0.f ? v : NEG_SLOPE * v;
    atomicMax(&m[d * HEADS + h], ordmap(v));
  }
}

// ---------------- edge pass B: ee = exp(e - m[dst]); z += ee ----------------
__global__ void edge_expsum(const int* __restrict__ src, const int* __restrict__ dst,
                            int E, int n, const float* __restrict__ al_s,
                            const float* __restrict__ al_d,
                            const unsigned* __restrict__ m,
                            float* __restrict__ ee, float* __restrict__ z) {
  int e = blockIdx.x * blockDim.x + threadIdx.x;
  if (e >= E + n) return;
  int s, d;
  edge_sd(src, dst, E, e, s, d);
#pragma unroll
  for (int h = 0; h < HEADS; ++h) {
    float v = al_s[s * HEADS + h] + al_d[d * HEADS + h];
    v = v > 0.f ? v : NEG_SLOPE * v;
    float x = expf(v - ordunmap(m[d * HEADS + h]));
    ee[(size_t)e * HEADS + h] = x;
    atomicAdd(&z[d * HEADS + h], x);
  }
}

// -------- edge pass C: out[dst,h,:] += H[src,h,:] * (ee / z[dst,h]) ---------
__global__ void edge_message(const int* __restrict__ src, const int* __restrict__ dst,
                             int E, int n, const float* __restrict__ H,
                             const float* __restrict__ ee,
                             const float* __restrict__ z,
                             float* __restrict__ out) {
  long long i = (long long)blockIdx.x * blockDim.x + threadIdx.x;  // (edge,head)
  long long total = (long long)(E + n) * HEADS;
  if (i >= total) return;
  int e = (int)(i >> 3);
  int h = (int)(i & 7);
  int s, d;
  edge_sd(src, dst, E, e, s, d);
  float alpha = ee[i] / z[d * HEADS + h];
  // 64B-aligned 16-float row: gather as 4x global_load_b128
  const float4* __restrict__ hp4 =
      (const float4*)(H + (size_t)s * FDIM + h * HID);
  float* op = out + (size_t)d * FDIM + h * HID;
#pragma unroll
  for (int q = 0; q < 4; ++q) {
    float4 v = hp4[q];
    atomicAdd(&op[q * 4 + 0], v.x * alpha);
    atomicAdd(&op[q * 4 + 1], v.y * alpha);
    atomicAdd(&op[q * 4 + 2], v.z * alpha);
    atomicAdd(&op[q * 4 + 3], v.w * alpha);
  }
}

// ---------------- bias + ELU (in place OK) ----------------------------------
__global__ void bias_elu(const float* __restrict__ in, const float* __restrict__ b,
                         float* __restrict__ out, size_t total) {
  size_t i = (size_t)blockIdx.x * blockDim.x + threadIdx.x;
  if (i >= total) return;
  float v = in[i] + b[i % FDIM];
  out[i] = v > 0.f ? v : (expf(v) - 1.0f);
}

// ---------------- head mean + bias ------------------------------------------
__global__ void mean_bias(const float* __restrict__ acc, const float* __restrict__ b2,
                          float* __restrict__ y, int n) {
  int i = blockIdx.x * blockDim.x + threadIdx.x;  // over n*CLS
  if (i >= n * CLS) return;
  int node = i / CLS, c = i % CLS;
  float s = 0.f;
#pragma unroll
  for (int h = 0; h < HEADS; ++h) s += acc[(size_t)node * FDIM + h * CLS + c];
  y[i] = s * (1.0f / HEADS) + b2[c];
}

// ---------------- GCN propagation -------------------------------------------
__global__ void deg_count(const int* __restrict__ dst, int E, int n,
                          float* __restrict__ deg) {
  int e = blockIdx.x * blockDim.x + threadIdx.x;
  if (e >= E + n) return;
  int d = (e < E) ? dst[e] : (e - E);
  atomicAdd(&deg[d], 1.0f);
}
__global__ void deg_inv(float* __restrict__ deg, int n) {
  int i = blockIdx.x * blockDim.x + threadIdx.x;
  if (i >= n) return;
  float d = deg[i];
  deg[i] = d > 0.f ? 1.0f / sqrtf(d) : 0.0f;
}
__global__ void iconv_scatter(const int* __restrict__ src, const int* __restrict__ dst,
                              int E, int n, const float* __restrict__ y,
                              const float* __restrict__ dinv,
                              float* __restrict__ out) {
  int e = blockIdx.x * blockDim.x + threadIdx.x;
  if (e >= E + n) return;
  int s, d;
  edge_sd(src, dst, E, e, s, d);
  float w = dinv[s] * dinv[d];
  const float4* __restrict__ yp4 = (const float4*)(y + (size_t)s * CLS);
  float* op = out + (size_t)d * CLS;
#pragma unroll
  for (int q = 0; q < 4; ++q) {
    float4 v = yp4[q];
    atomicAdd(&op[q * 4 + 0], v.x * w);
    atomicAdd(&op[q * 4 + 1], v.y * w);
    atomicAdd(&op[q * 4 + 2], v.z * w);
    atomicAdd(&op[q * 4 + 3], v.w * w);
  }
}

// ---------------------------------------------------------------------------
static inline int cdiv(long long a, int b) { return (int)((a + b - 1) / b); }

extern "C" void kernel_launch(void* const* d_in, const int* in_sizes, int n_in,
                              void* d_out, int out_size, void* d_ws, size_t ws_size,
                              hipStream_t stream) {
  const float* x      = (const float*)d_in[0];
  const int*   ei     = (const int*)d_in[1];
  const float* W1     = (const float*)d_in[2];
  const float* a_src1 = (const float*)d_in[3];
  const float* a_dst1 = (const float*)d_in[4];
  const float* b1     = (const float*)d_in[5];
  const float* W2     = (const float*)d_in[6];
  const float* a_src2 = (const float*)d_in[7];
  const float* a_dst2 = (const float*)d_in[8];
  const float* b2     = (const float*)d_in[9];
  float* out = (float*)d_out;

  const int n = in_sizes[0] / FDIM;     // 50000
  const int E = in_sizes[1] / 2;        // 800000
  const int ET = E + n;                 // with self loops
  const int* src = ei;
  const int* dst = ei + E;

  // workspace carve-up (floats)
  float* ws = (float*)d_ws;
  const size_t NF = (size_t)n * FDIM;
  float*    h_buf = ws;                               // h1, later h2
  float*    g_buf = ws + NF;                          // out1/g1, later out2 accum
  float*    al_s  = ws + 2 * NF;
  float*    al_d  = al_s + (size_t)n * HEADS;
  unsigned* mbuf  = (unsigned*)(al_d + (size_t)n * HEADS);
  float*    zbuf  = (float*)(mbuf + (size_t)n * HEADS);
  float*    eebuf = zbuf + (size_t)n * HEADS;
  float*    deg   = eebuf + (size_t)ET * HEADS;
  float*    ybuf  = deg + n;

  const int TB = 256;
  const int gemmBlocks = cdiv(n / 16, 8);             // 8 row-tiles (waves) per block
  const unsigned MININF = ORDMAP_NEG_INF;

  // ---------------- layer 1 ----------------
  gemm128_wmma<<<gemmBlocks, TB, 0, stream>>>(x, W1, h_buf, n);
  attn_logits<<<cdiv((long long)n * HEADS, TB), TB, 0, stream>>>(
      h_buf, a_src1, a_dst1, al_s, al_d, n);
  fill_u32<<<cdiv((long long)n * HEADS, TB), TB, 0, stream>>>(mbuf, MININF, (size_t)n * HEADS);
  fill_f32<<<cdiv((long long)n * HEADS, TB), TB, 0, stream>>>(zbuf, 0.f, (size_t)n * HEADS);
  fill_f32<<<cdiv((long long)NF, TB), TB, 0, stream>>>(g_buf, 0.f, NF);
  edge_max<<<cdiv(ET, TB), TB, 0, stream>>>(src, dst, E, n, al_s, al_d, mbuf);
  edge_expsum<<<cdiv(ET, TB), TB, 0, stream>>>(src, dst, E, n, al_s, al_d, mbuf, eebuf, zbuf);
  edge_message<<<cdiv((long long)ET * HEADS, TB), TB, 0, stream>>>(
      src, dst, E, n, h_buf, eebuf, zbuf, g_buf);
  bias_elu<<<cdiv((long long)NF, TB), TB, 0, stream>>>(g_buf, b1, g_buf, NF);

  // ---------------- layer 2 ----------------
  gemm128_wmma<<<gemmBlocks, TB, 0, stream>>>(g_buf, W2, h_buf, n);
  attn_logits<<<cdiv((long long)n * HEADS, TB), TB, 0, stream>>>(
      h_buf, a_src2, a_dst2, al_s, al_d, n);
  fill_u32<<<cdiv((long long)n * HEADS, TB), TB, 0, stream>>>(mbuf, MININF, (size_t)n * HEADS);
  fill_f32<<<cdiv((long long)n * HEADS, TB), TB, 0, stream>>>(zbuf, 0.f, (size_t)n * HEADS);
  fill_f32<<<cdiv((long long)NF, TB), TB, 0, stream>>>(g_buf, 0.f, NF);  // reuse as out2 accum
  edge_max<<<cdiv(ET, TB), TB, 0, stream>>>(src, dst, E, n, al_s, al_d, mbuf);
  edge_expsum<<<cdiv(ET, TB), TB, 0, stream>>>(src, dst, E, n, al_s, al_d, mbuf, eebuf, zbuf);
  edge_message<<<cdiv((long long)ET * HEADS, TB), TB, 0, stream>>>(
      src, dst, E, n, h_buf, eebuf, zbuf, g_buf);
  mean_bias<<<cdiv((long long)n * CLS, TB), TB, 0, stream>>>(g_buf, b2, ybuf, n);

  // ---------------- GCN propagation ----------------
  fill_f32<<<cdiv(n, TB), TB, 0, stream>>>(deg, 0.f, (size_t)n);
  fill_f32<<<cdiv((long long)n * CLS, TB), TB, 0, stream>>>(out, 0.f, (size_t)n * CLS);
  deg_count<<<cdiv(ET, TB), TB, 0, stream>>>(dst, E, n, deg);
  deg_inv<<<cdiv(n, TB), TB, 0, stream>>>(deg, n);
  iconv_scatter<<<cdiv(ET, TB), TB, 0, stream>>>(src, dst, E, n, ybuf, deg, out);
}